// BindingConstraintsAlphaBetaN_137438954250
// MI455X (gfx1250) — compile-verified
//
#include <hip/hip_runtime.h>
#include <hip/hip_bf16.h>
#include <math.h>

// ---------------- problem constants (match reference) ----------------
#define NB        16
#define NFRAG     3000
#define APF       3
#define NATM      (NB * NFRAG * APF)      // 144000
#define DIM       256
#define ROWS_PER_MOL (NFRAG * APF)        // 9000
#define D2_AB     (1.5f * 1.5f)
#define D2_AN     (2.4f * 2.4f)
#define N_ITER    10
#define CONVERGED 1e-4f

typedef float v2f __attribute__((ext_vector_type(2)));
typedef float v8f __attribute__((ext_vector_type(8)));
typedef unsigned int u32x4 __attribute__((ext_vector_type(4)));
typedef int i32x4 __attribute__((ext_vector_type(4)));
typedef int i32x8 __attribute__((ext_vector_type(8)));

// ---------------- workspace layout (float offsets into d_ws) ----------------
#define WS_M      0        // 9 floats : M = U3 @ P3
#define WS_S      16       // 9 floats : S = U3 @ U3^T
#define WS_CNORM  32
#define WS_LAMSQ  33
#define WS_ALPHA  34
#define WS_STEP   35
#define WS_DONE   36       // int stored in this slot
#define WS_A0     64       // 32 floats  (Σ_f c)
#define WS_B1     96       // 32 floats  (Σ_f dx·gdx)
#define WS_C2     128      // 32 floats  (Σ_f |gdx|²)
#define WS_X3     1024     // NATM*3 floats
#define WS_GACC   (WS_X3 + NATM * 3)

// ---------------- TDM helpers ----------------------------------------------
// LDS byte offset of a __shared__ object (generic -> addrspace(3) cast).
__device__ __forceinline__ uint32_t lds_addr_u32(const void* p) {
  return (uint32_t)(uintptr_t)(const __attribute__((address_space(3))) char*)p;
}

// Issue TENSOR_LOAD_TO_LDS for a 2D f32 tile: `rows` rows of `cols` elements,
// global row stride `row_stride_elems`, optional LDS padding (pad_amount
// DWORD-code inserted every pad_interval DWORD-code) for bank-conflict-free
// consumption. D# layout per CDNA5 ISA §8 (group0 + group1; groups 2/3 NULL).
__device__ __forceinline__ void tdm_load_f32_2d(
    uint32_t lds_byte, const void* gsrc,
    uint32_t cols, uint32_t rows, uint32_t row_stride_elems,
    uint32_t pad_en, uint32_t pad_interval_code, uint32_t pad_amount_code) {
  uint64_t ga = (uint64_t)(uintptr_t)gsrc;
  u32x4 g0;
  g0[0] = 1u;                                     // count=1 (valid user D#)
  g0[1] = lds_byte;                               // lds_addr  (bits 63:32)
  g0[2] = (uint32_t)ga;                           // global_addr[31:0]
  g0[3] = (uint32_t)((ga >> 32) & 0x01ffffffu)    // global_addr[56:32]
        | (2u << 30);                             // type=2 ("image")
  i32x8 g1;
  g1[0] = (int)((2u << 16)                        // data_size = 4 bytes
              | (pad_en << 20)
              | (pad_interval_code << 22)
              | (pad_amount_code << 25));
  g1[1] = (int)((cols & 0xffffu) << 16);          // tensor_dim0[15:0]
  g1[2] = (int)((cols >> 16) | ((rows & 0xffffu) << 16)); // td0 hi | td1 lo
  g1[3] = (int)((rows >> 16) | ((cols & 0xffffu) << 16)); // td1 hi | tile_dim0
  g1[4] = (int)(rows & 0xffffu);                  // tile_dim1 (tile_dim2=0)
  g1[5] = (int)row_stride_elems;                  // tensor_dim0_stride[31:0]
  g1[6] = 0;
  g1[7] = 0;
  i32x4 z4 = {0, 0, 0, 0};
#if defined(__clang_major__) && (__clang_major__ >= 23)
  i32x8 z8 = {0, 0, 0, 0, 0, 0, 0, 0};
  __builtin_amdgcn_tensor_load_to_lds(g0, g1, z4, z4, z8, 0);
#else
  __builtin_amdgcn_tensor_load_to_lds(g0, g1, z4, z4, 0);
#endif
}

// ---------------- init: zero accumulators, build 3x3 matrices -------------
__global__ void init_kernel(const float* __restrict__ P, const float* __restrict__ U,
                            float* __restrict__ ws) {
  size_t tid    = (size_t)blockIdx.x * blockDim.x + threadIdx.x;
  size_t stride = (size_t)gridDim.x * blockDim.x;
  for (size_t i = tid; i < (size_t)NATM * 3; i += stride)
    ws[WS_GACC + i] = 0.0f;

  if (blockIdx.x == 0) {
    int t = threadIdx.x;
    if (t < 9) {                       // M[i][j] = Σ_k U[i,k] * P[k,j]
      int i = t / 3, j = t % 3;
      float acc = 0.f;
      for (int k = 0; k < DIM; ++k) acc += U[i * DIM + k] * P[k * 6 + j];
      ws[WS_M + t] = acc;
    } else if (t >= 16 && t < 25) {    // S[i][j] = Σ_k U[i,k] * U[j,k]
      int tt = t - 16, i = tt / 3, j = tt % 3;
      float acc = 0.f;
      for (int k = 0; k < DIM; ++k) acc += U[i * DIM + k] * U[j * DIM + k];
      ws[WS_S + tt] = acc;
    }
    if (t == 32) {
      ws[WS_CNORM] = 0.f; ws[WS_LAMSQ] = 0.f;
      ws[WS_ALPHA] = 0.f; ws[WS_STEP]  = 0.f;
      ((int*)ws)[WS_DONE] = 0;
    }
    if (t >= 64 && t < 160) ws[t] = 0.f;   // A0, B1, C2
  }
}

// ---------------- projection: x3 = y @ P[:, :3] -----------------------------
// TDM stages each wave's 16x256 f32 tile to LDS (4-DWORD row pad ->
// 260-DWORD stride, bank-conflict-free A reads), then 64 x
// V_WMMA_F32_16X16X4_F32 over K=256. B fragments precomputed in LDS.
#define PRJ_WAVES  8
#define PRJ_STRIDE 260     // 256 cols + 4 pad DWORDs per row
__global__ void __launch_bounds__(256)
project_wmma(const float* __restrict__ y, const float* __restrict__ P,
             float* __restrict__ ws) {
  __shared__ float ytile[PRJ_WAVES * 16 * PRJ_STRIDE];  // 133,120 B
  __shared__ float Bfrag[64 * 64];                      // 16,384 B [step][lane*2+j]
  const int wave = threadIdx.x >> 5;
  const int lane = threadIdx.x & 31;
  const int tile = blockIdx.x * PRJ_WAVES + wave;       // grid exact: 1125*8=9000
  const bool hi  = lane >= 16;
  const int  lc  = lane & 15;
  float* myTile = &ytile[wave * 16 * PRJ_STRIDE];

  // async DMA: 16 rows x 256 cols, contiguous rows, pad 4 DW every 256 DW
  tdm_load_f32_2d(lds_addr_u32(myTile), y + (size_t)tile * 16 * DIM,
                  DIM, 16, DIM, /*pad*/1, /*interval=256DW*/7, /*amount=4DW*/3);

  // cooperative fill of B fragments (P3 padded 3 -> 16 columns)
  for (int i = 0; i < 16; ++i) {
    int e = threadIdx.x * 16 + i;            // 4096 entries
    int step = e >> 6, r = e & 63;
    int l = r >> 1, j = r & 1;
    int elc = l & 15;
    int K = step * 4 + ((l >= 16) ? 2 : 0) + j;
    float pv = P[K * 6 + (elc < 3 ? elc : 0)];
    Bfrag[e] = (elc < 3) ? pv : 0.f;
  }
  __syncthreads();                           // Bfrag ready (all waves)
  __builtin_amdgcn_s_wait_tensorcnt(0);      // this wave's tile ready

  v8f acc = {0.f, 0.f, 0.f, 0.f, 0.f, 0.f, 0.f, 0.f};
  const float* arow = &myTile[lc * PRJ_STRIDE + (hi ? 2 : 0)];
  for (int kb = 0; kb < DIM; kb += 4) {
    v2f a = *(const v2f*)&arow[kb];                        // ds_load_b64
    v2f b = *(const v2f*)&Bfrag[(kb >> 2) * 64 + lane * 2]; // ds_load_b64
    acc = __builtin_amdgcn_wmma_f32_16x16x4_f32(false, a, false, b,
                                                (short)0, acc, false, false);
  }
  const int rowBase = tile * 16;
  for (int v = 0; v < 8; ++v) {
    int row = rowBase + v + (hi ? 8 : 0);
    if (lc < 3) ws[WS_X3 + row * 3 + lc] = acc[v];
  }
}

__device__ __forceinline__ float quadS(const float* S, float gx, float gy, float gz) {
  return S[0]*gx*gx + S[4]*gy*gy + S[8]*gz*gz
       + 2.f*(S[1]*gx*gy + S[2]*gx*gz + S[5]*gy*gz);
}

// ---------------- per-iteration reductions ---------------------------------
__global__ void reduce_kernel(float* __restrict__ ws) {
  __shared__ float sA0[32], sB1[32], sC2[32], sCn, sLam;
  int t = threadIdx.x;
  if (t < 32) { sA0[t] = 0.f; sB1[t] = 0.f; sC2[t] = 0.f; }
  if (t == 32) { sCn = 0.f; sLam = 0.f; }
  __syncthreads();

  float Mm[9], Ss[9];
  for (int i = 0; i < 9; ++i) { Mm[i] = ws[WS_M + i]; Ss[i] = ws[WS_S + i]; }

  int f = blockIdx.x * blockDim.x + t;
  if (f < NFRAG) {
    float fragsq = 0.f, lamloc = 0.f;
    for (int b = 0; b < NB; ++b) {
      const float* r0 = ws + WS_X3 + ((size_t)b * ROWS_PER_MOL + 3 * f) * 3;
      float ax=r0[0], ay=r0[1], az=r0[2];
      float bx=r0[3], by=r0[4], bz=r0[5];
      float nx=r0[6], ny=r0[7], nz=r0[8];
      float d0x=bx-ax, d0y=by-ay, d0z=bz-az;
      float d1x=nx-ax, d1y=ny-ay, d1z=nz-az;
      float c0 = d0x*d0x + d0y*d0y + d0z*d0z - D2_AB;
      float c1 = d1x*d1x + d1y*d1y + d1z*d1z - D2_AN;
      fragsq += c0*c0 + c1*c1;
      float C0x=c0*d0x, C0y=c0*d0y, C0z=c0*d0z;
      float C1x=c1*d1x, C1y=c1*d1y, C1z=c1*d1z;
      float gax=-2.f*(C0x+C1x), gay=-2.f*(C0y+C1y), gaz=-2.f*(C0z+C1z);
      float gbx= 2.f*C0x,       gby= 2.f*C0y,       gbz= 2.f*C0z;
      float gnx= 2.f*C1x,       gny= 2.f*C1y,       gnz= 2.f*C1z;
      lamloc += quadS(Ss,gax,gay,gaz) + quadS(Ss,gbx,gby,gbz) + quadS(Ss,gnx,gny,gnz);
      float e0x=gbx-gax, e0y=gby-gay, e0z=gbz-gaz;
      float e1x=gnx-gax, e1y=gny-gay, e1z=gnz-gaz;
      float h0x=e0x*Mm[0]+e0y*Mm[3]+e0z*Mm[6];
      float h0y=e0x*Mm[1]+e0y*Mm[4]+e0z*Mm[7];
      float h0z=e0x*Mm[2]+e0y*Mm[5]+e0z*Mm[8];
      float h1x=e1x*Mm[0]+e1y*Mm[3]+e1z*Mm[6];
      float h1y=e1x*Mm[1]+e1y*Mm[4]+e1z*Mm[7];
      float h1z=e1x*Mm[2]+e1y*Mm[5]+e1z*Mm[8];
      atomicAdd(&sA0[b*2+0], c0);
      atomicAdd(&sA0[b*2+1], c1);
      atomicAdd(&sB1[b*2+0], d0x*h0x + d0y*h0y + d0z*h0z);
      atomicAdd(&sB1[b*2+1], d1x*h1x + d1y*h1y + d1z*h1z);
      atomicAdd(&sC2[b*2+0], h0x*h0x + h0y*h0y + h0z*h0z);
      atomicAdd(&sC2[b*2+1], h1x*h1x + h1y*h1y + h1z*h1z);
    }
    atomicAdd(&sCn,  sqrtf(fragsq));   // Σ_f sqrt(Σ_{b,k} c²)
    atomicAdd(&sLam, lamloc);
  }
  __syncthreads();
  if (t < 32) {
    atomicAdd(&ws[WS_A0 + t], sA0[t]);
    atomicAdd(&ws[WS_B1 + t], sB1[t]);
    atomicAdd(&ws[WS_C2 + t], sC2[t]);
  }
  if (t == 32) { atomicAdd(&ws[WS_CNORM], sCn); atomicAdd(&ws[WS_LAMSQ], sLam); }
}

// ---------------- scalar line search: ctry(a) = A0 - 2a*B1 + a^2*C2 --------
__global__ void linesearch_kernel(float* __restrict__ ws, int iter) {
  if (threadIdx.x != 0 || blockIdx.x != 0) return;
  float cnorm = ws[WS_CNORM];
  float alpha = (iter == 0) ? (1.0f / sqrtf(ws[WS_LAMSQ])) : ws[WS_ALPHA];
  float A0[32], B1[32], C2[32];
  for (int i = 0; i < 32; ++i) {
    A0[i] = ws[WS_A0 + i]; B1[i] = ws[WS_B1 + i]; C2[i] = ws[WS_C2 + i];
  }
  int ls = 0; float cn = 0.f; bool dn = false;
  while (!dn) {
    float a = alpha, a2 = a * a, s = 0.f;
    for (int i = 0; i < 32; ++i) {
      float t = A0[i] - 2.f * a * B1[i] + a2 * C2[i];
      s += t * t;
    }
    cn = sqrtf(s);
    bool accept = cn < cnorm;
    if (!accept) { alpha *= 0.5f; ls += 1; }
    dn = accept || (ls > 10);
  }
  if (ls == 0 && cn > CONVERGED) alpha *= 1.5f;

  int done_old = ((int*)ws)[WS_DONE];
  ws[WS_STEP] = done_old ? 0.f : alpha;
  if (!done_old) ws[WS_ALPHA] = alpha;
  ((int*)ws)[WS_DONE] = done_old | (cn < CONVERGED ? 1 : 0);

  ws[WS_CNORM] = 0.f; ws[WS_LAMSQ] = 0.f;
  for (int i = 0; i < 32; ++i) { ws[WS_A0+i]=0.f; ws[WS_B1+i]=0.f; ws[WS_C2+i]=0.f; }
}

// ---------------- apply step: x3 -= step * grad@M ; Gacc += step * grad ----
__global__ void update_kernel(float* __restrict__ ws) {
  int idx = blockIdx.x * blockDim.x + threadIdx.x;
  if (idx >= NB * NFRAG) return;
  float step = ws[WS_STEP];
  int b = idx / NFRAG, f = idx % NFRAG;
  float Mm[9];
  for (int i = 0; i < 9; ++i) Mm[i] = ws[WS_M + i];
  float* r0 = ws + WS_X3   + ((size_t)b * ROWS_PER_MOL + 3 * f) * 3;
  float* g0 = ws + WS_GACC + ((size_t)b * ROWS_PER_MOL + 3 * f) * 3;

  float ax=r0[0], ay=r0[1], az=r0[2];
  float bx=r0[3], by=r0[4], bz=r0[5];
  float nx=r0[6], ny=r0[7], nz=r0[8];
  float d0x=bx-ax, d0y=by-ay, d0z=bz-az;
  float d1x=nx-ax, d1y=ny-ay, d1z=nz-az;
  float c0 = d0x*d0x + d0y*d0y + d0z*d0z - D2_AB;
  float c1 = d1x*d1x + d1y*d1y + d1z*d1z - D2_AN;
  float C0x=c0*d0x, C0y=c0*d0y, C0z=c0*d0z;
  float C1x=c1*d1x, C1y=c1*d1y, C1z=c1*d1z;
  float g[9] = { -2.f*(C0x+C1x), -2.f*(C0y+C1y), -2.f*(C0z+C1z),
                  2.f*C0x,        2.f*C0y,        2.f*C0z,
                  2.f*C1x,        2.f*C1y,        2.f*C1z };
  for (int a = 0; a < 3; ++a) {
    float gx = g[a*3+0], gy = g[a*3+1], gz = g[a*3+2];
    r0[a*3+0] -= step * (gx*Mm[0] + gy*Mm[3] + gz*Mm[6]);
    r0[a*3+1] -= step * (gx*Mm[1] + gy*Mm[4] + gz*Mm[7]);
    r0[a*3+2] -= step * (gx*Mm[2] + gy*Mm[5] + gz*Mm[8]);
    g0[a*3+0] += step * gx;
    g0[a*3+1] += step * gy;
    g0[a*3+2] += step * gz;
  }
}

// ---------------- uplift: out = y - Gacc @ U3 -------------------------------
// Wave0 TDM-stages the block's 16x128 y tile (strided rows, padded in LDS),
// then each wave does one V_WMMA_F32_16X16X4_F32: D = (-Gacc)*U3 + y.
#define UPL_STRIDE 132     // 128 cols + 4 pad DWORDs per row
__global__ void __launch_bounds__(256)
uplift_wmma(const float* __restrict__ y, const float* __restrict__ U,
            const float* __restrict__ ws, float* __restrict__ out) {
  __shared__ float ylds[16 * UPL_STRIDE];               // 8,448 B
  const int wave = threadIdx.x >> 5;
  const int lane = threadIdx.x & 31;
  const int tile = blockIdx.x * 8 + wave;               // grid exact: 18000*8
  const int rowBase = (tile >> 4) * 16;                 // uniform per block
  const int colBase = (tile & 15) * 16;
  const bool hi = lane >= 16;
  const int  lc = lane & 15;
  const float* G = ws + WS_GACC;

  if (wave == 0) {
    int blkColBase = (blockIdx.x & 1) * 128;            // block's 128-col slab
    tdm_load_f32_2d(lds_addr_u32(ylds),
                    y + (size_t)rowBase * DIM + blkColBase,
                    128, 16, DIM, /*pad*/1, /*interval=128DW*/6, /*amount=4DW*/3);
    __builtin_amdgcn_s_wait_tensorcnt(0);
  }

  v2f a;                                  // A = -Gacc (16x4, K=3 zero-padded)
  {
    int ar = rowBase + lc;
    __builtin_prefetch(&G[(size_t)ar * 3], 0, 1);
    if (!hi) { a.x = -G[(size_t)ar*3+0]; a.y = -G[(size_t)ar*3+1]; }
    else     { a.x = -G[(size_t)ar*3+2]; a.y = 0.f; }
  }
  v2f b;                                  // B = U3 (4x16, row 3 zero)
  {
    int bc = colBase + lc;
    if (!hi) { b.x = U[0*DIM + bc]; b.y = U[1*DIM + bc]; }
    else     { b.x = U[2*DIM + bc]; b.y = 0.f; }
  }
  __syncthreads();                        // staged tile visible to all waves

  v8f c;                                  // C = y tile from LDS (conflict-free)
  const int lcol = wave * 16 + lc;
  for (int v = 0; v < 8; ++v)
    c[v] = ylds[(v + (hi ? 8 : 0)) * UPL_STRIDE + lcol];

  v8f d = __builtin_amdgcn_wmma_f32_16x16x4_f32(false, a, false, b,
                                                (short)0, c, false, false);
  for (int v = 0; v < 8; ++v) {
    int row = rowBase + v + (hi ? 8 : 0);
    out[(size_t)row * DIM + colBase + lc] = d[v];
  }
}

// ---------------- host-side launch sequence --------------------------------
extern "C" void kernel_launch(void* const* d_in, const int* in_sizes, int n_in,
                              void* d_out, int out_size, void* d_ws, size_t ws_size,
                              hipStream_t stream) {
  const float* y = (const float*)d_in[0];   // [NATM, DIM]
  const float* P = (const float*)d_in[1];   // [DIM, 6]
  const float* U = (const float*)d_in[2];   // [6, DIM]
  float* ws  = (float*)d_ws;
  float* out = (float*)d_out;

  init_kernel<<<(NATM * 3 + 255) / 256, 256, 0, stream>>>(P, U, ws);

  // x3 = y @ P3 : 9000 row tiles, 8 waves/block, TDM-staged
  project_wmma<<<(NATM / 16) / PRJ_WAVES, 256, 0, stream>>>(y, P, ws);

  for (int j = 0; j < N_ITER; ++j) {
    reduce_kernel<<<(NFRAG + 255) / 256, 256, 0, stream>>>(ws);
    linesearch_kernel<<<1, 1, 0, stream>>>(ws, j);
    update_kernel<<<(NB * NFRAG + 255) / 256, 256, 0, stream>>>(ws);
  }

  // out = y - Gacc @ U3 : 144000 tiles, 8 per block, TDM-staged C operand
  uplift_wmma<<<((NATM / 16) * (DIM / 16)) / 8, 256, 0, stream>>>(y, U, ws, out);
}